// HierarchyPrototypeClassifier_10892037062678
// MI455X (gfx1250) — compile-verified
//
#include <hip/hip_runtime.h>

typedef __attribute__((ext_vector_type(2))) float v2f;
typedef __attribute__((ext_vector_type(8))) float v8f;

#define BATCH   32768
#define LATENT  512
#define NSUP    256
#define NSUB    1024
#define OUTN    200
#define NC      400          // fused output columns: [0,200)=sup head, [200,400)=sub head
#define RB      64           // x rows per block in main kernel
#define LPAD    516          // padded LDS row stride (floats) -> conflict-free A fragments
#define FLT_BIG 3.402823466e38f

// order-preserving float<->uint mapping so we can use integer atomicMin
__device__ __forceinline__ unsigned fflip(float f) {
  unsigned u = __float_as_uint(f);
  return (u & 0x80000000u) ? ~u : (u | 0x80000000u);
}
__device__ __forceinline__ float funflip(unsigned u) {
  u = (u & 0x80000000u) ? (u & 0x7FFFFFFFu) : ~u;
  return __uint_as_float(u);
}

// ---- prelude: row sum-of-squares for prototypes ----
__global__ void k_row_sumsq(const float* __restrict__ sup, const float* __restrict__ sub,
                            float* __restrict__ yysup, float* __restrict__ yysub) {
  int t = blockIdx.x * blockDim.x + threadIdx.x;
  if (t >= NSUP + NSUB) return;
  const float* row = (t < NSUP) ? (sup + (size_t)t * LATENT)
                                : (sub + (size_t)(t - NSUP) * LATENT);
  const float4* r4 = (const float4*)row;
  float s = 0.f;
  for (int k = 0; k < LATENT / 4; ++k) {
    float4 v = r4[k];
    s += v.x * v.x + v.y * v.y + v.z * v.z + v.w * v.w;
  }
  if (t < NSUP) yysup[t] = s; else yysub[t - NSUP] = s;
}

// ---- prelude: init global colmin + r2 accumulator ----
__global__ void k_prep(unsigned* __restrict__ colmin, float* __restrict__ r2part) {
  int t = blockIdx.x * blockDim.x + threadIdx.x;
  if (t < NSUB) colmin[t] = 0xFFFFFFFFu;
  if (t == 0) *r2part = 0.f;
}

// ---- prelude: per-output-column scale (rowsum W) and constant (yy.W + b) ----
__global__ void k_svec_cvec(const float* __restrict__ W1, const float* __restrict__ b1,
                            const float* __restrict__ W2, const float* __restrict__ b2,
                            const float* __restrict__ yysup, const float* __restrict__ yysub,
                            float* __restrict__ svec, float* __restrict__ cvec) {
  int n = blockIdx.x * blockDim.x + threadIdx.x;
  if (n >= NC) return;
  float s = 0.f, c = 0.f;
  if (n < OUTN) {
    const float* w = W1 + (size_t)n * NSUP;
    for (int j = 0; j < NSUP; ++j) { s += w[j]; c += yysup[j] * w[j]; }
    c += b1[n];
  } else {
    const float* w = W2 + (size_t)(n - OUTN) * NSUB;
    for (int j = 0; j < NSUB; ++j) { s += w[j]; c += yysub[j] * w[j]; }
    c += b2[n - OUTN];
  }
  svec[n] = s;
  cvec[n] = c;
}

// ---- prelude: fused head matrix, n-major: M[n][k] = sum_j W[n][j] * Proto[j][k] ----
// (same layout as Sub rows so pass-2 B fragments are contiguous b64 loads)
__global__ void k_build_M(const float* __restrict__ sup, const float* __restrict__ sub,
                          const float* __restrict__ W1, const float* __restrict__ W2,
                          float* __restrict__ M) {
  int t = blockIdx.x * blockDim.x + threadIdx.x;
  if (t >= LATENT * NC) return;
  int n = t / LATENT, k = t % LATENT;     // consecutive threads -> consecutive k (coalesced)
  float acc = 0.f;
  if (n < OUTN) {
    const float* w = W1 + (size_t)n * NSUP;
    for (int j = 0; j < NSUP; ++j) acc += w[j] * sup[(size_t)j * LATENT + k];
  } else {
    const float* w = W2 + (size_t)(n - OUTN) * NSUB;
    for (int j = 0; j < NSUB; ++j) acc += w[j] * sub[(size_t)j * LATENT + k];
  }
  M[(size_t)n * LATENT + k] = acc;
}

// ---- prelude: super-sub prototype distance matrix (256 x 1024, tiny: 0.27 GF) ----
__global__ void k_dss(const float* __restrict__ sup, const float* __restrict__ sub,
                      const float* __restrict__ yysup, const float* __restrict__ yysub,
                      float* __restrict__ D) {
  int t = blockIdx.x * blockDim.x + threadIdx.x;
  if (t >= NSUP * NSUB) return;
  int i = t / NSUB, j = t % NSUB;
  const float4* a = (const float4*)(sup + (size_t)i * LATENT);
  const float4* b = (const float4*)(sub + (size_t)j * LATENT);
  float d = 0.f;
  for (int k = 0; k < LATENT / 4; ++k) {
    float4 va = a[k], vb = b[k];
    d += va.x * vb.x + va.y * vb.y + va.z * vb.z + va.w * vb.w;
  }
  D[t] = yysup[i] - 2.f * d + yysub[j];
}

__global__ void k_r3(const float* __restrict__ D, float* __restrict__ out_r3) {
  __shared__ float red[256];
  int t = threadIdx.x;
  const float* row = D + (size_t)t * NSUB;
  float m = FLT_BIG;
  for (int j = 0; j < NSUB; ++j) m = fminf(m, row[j]);
  red[t] = m;
  __syncthreads();
  for (int s = 128; s > 0; s >>= 1) {
    if (t < s) red[t] += red[t + s];
    __syncthreads();
  }
  if (t == 0) out_r3[0] = red[0] / (float)NSUP;
}

__global__ void k_r4(const float* __restrict__ D, float* __restrict__ out_r4) {
  __shared__ float red[1024];
  int t = threadIdx.x;
  float m = FLT_BIG;
  for (int i = 0; i < NSUP; ++i) m = fminf(m, D[(size_t)i * NSUB + t]);
  red[t] = m;
  __syncthreads();
  for (int s = 512; s > 0; s >>= 1) {
    if (t < s) red[t] += red[t + s];
    __syncthreads();
  }
  if (t == 0) out_r4[0] = red[0] / (float)NSUB;
}

// ---- main fused kernel: WMMA f32 distance GEMM (+min fusion) and WMMA head GEMM ----
__global__ __launch_bounds__(256) void k_main(
    const float* __restrict__ x, const float* __restrict__ sub,
    const float* __restrict__ yysub, const float* __restrict__ Mm,
    const float* __restrict__ svec, const float* __restrict__ cvec,
    float* __restrict__ out_sub, float* __restrict__ out_sup,
    unsigned* __restrict__ g_colmin, float* __restrict__ g_r2part) {
  extern __shared__ float sm[];
  float*    Xs      = sm;                       // RB * LPAD
  float*    xxs     = sm + RB * LPAD;           // RB
  unsigned* rowminU = (unsigned*)(xxs + RB);    // RB
  unsigned* colminU = rowminU + RB;             // NSUB

  const int tid  = threadIdx.x;
  const int row0 = blockIdx.x * RB;

  // stage 64x512 x-block into LDS (padded rows)
  for (int idx = tid; idx < RB * (LATENT / 4); idx += 256) {
    const int r  = idx >> 7;
    const int c4 = (idx & 127) << 2;
    const float4 v = *(const float4*)(x + (size_t)(row0 + r) * LATENT + c4);
    float* d = Xs + r * LPAD + c4;
    d[0] = v.x; d[1] = v.y; d[2] = v.z; d[3] = v.w;
  }
  if (tid < RB) rowminU[tid] = 0xFFFFFFFFu;
  for (int j = tid; j < NSUB; j += 256) colminU[j] = 0xFFFFFFFFu;
  __syncthreads();
  if (tid < RB) {
    const float* rp = Xs + tid * LPAD;
    float s = 0.f;
    for (int k = 0; k < LATENT; ++k) s += rp[k] * rp[k];
    xxs[tid] = s;
  }
  __syncthreads();

  const int wave = tid >> 5;
  const int lane = tid & 31;
  const int half = lane >> 4;
  const int l16  = lane & 15;
  const v8f vzero = {0.f, 0.f, 0.f, 0.f, 0.f, 0.f, 0.f, 0.f};

  // ---- Pass 1: D = Xblk @ Sub^T (64 x 1024) with fused row/col min; never stored ----
  // supertile = 32 rows x 64 cols, 2x4 register blocking of 16x16x4 f32 WMMA
  for (int st = wave; st < 32; st += 8) {
    const int msup = (st & 1) * 32;
    const int nsup = (st >> 1) * 64;
    v8f acc[2][4];
#pragma unroll
    for (int mi = 0; mi < 2; ++mi)
#pragma unroll
      for (int ni = 0; ni < 4; ++ni) acc[mi][ni] = vzero;

    const float* a0p = Xs + (size_t)(msup + l16) * LPAD + 2 * half;
    const float* a1p = Xs + (size_t)(msup + 16 + l16) * LPAD + 2 * half;
    const float* b0p = sub + (size_t)(nsup +  0 + l16) * LATENT + 2 * half;
    const float* b1p = sub + (size_t)(nsup + 16 + l16) * LATENT + 2 * half;
    const float* b2p = sub + (size_t)(nsup + 32 + l16) * LATENT + 2 * half;
    const float* b3p = sub + (size_t)(nsup + 48 + l16) * LATENT + 2 * half;

    for (int kb = 0; kb < LATENT; kb += 4) {
      v2f a0 = *(const v2f*)(a0p + kb);
      v2f a1 = *(const v2f*)(a1p + kb);
      v2f b0 = *(const v2f*)(b0p + kb);
      v2f b1 = *(const v2f*)(b1p + kb);
      v2f b2 = *(const v2f*)(b2p + kb);
      v2f b3 = *(const v2f*)(b3p + kb);
      acc[0][0] = __builtin_amdgcn_wmma_f32_16x16x4_f32(false, a0, false, b0, (short)0, acc[0][0], false, false);
      acc[0][1] = __builtin_amdgcn_wmma_f32_16x16x4_f32(false, a0, false, b1, (short)0, acc[0][1], false, false);
      acc[0][2] = __builtin_amdgcn_wmma_f32_16x16x4_f32(false, a0, false, b2, (short)0, acc[0][2], false, false);
      acc[0][3] = __builtin_amdgcn_wmma_f32_16x16x4_f32(false, a0, false, b3, (short)0, acc[0][3], false, false);
      acc[1][0] = __builtin_amdgcn_wmma_f32_16x16x4_f32(false, a1, false, b0, (short)0, acc[1][0], false, false);
      acc[1][1] = __builtin_amdgcn_wmma_f32_16x16x4_f32(false, a1, false, b1, (short)0, acc[1][1], false, false);
      acc[1][2] = __builtin_amdgcn_wmma_f32_16x16x4_f32(false, a1, false, b2, (short)0, acc[1][2], false, false);
      acc[1][3] = __builtin_amdgcn_wmma_f32_16x16x4_f32(false, a1, false, b3, (short)0, acc[1][3], false, false);
    }

    // epilogue: dist = xx - 2*acc + yy ; fold row/col mins via flipped-uint atomics
    float xv[2][8];
#pragma unroll
    for (int mi = 0; mi < 2; ++mi)
#pragma unroll
      for (int r = 0; r < 8; ++r)
        xv[mi][r] = xxs[msup + mi * 16 + 8 * half + r];

    float rmn[2][8];
#pragma unroll
    for (int mi = 0; mi < 2; ++mi)
#pragma unroll
      for (int r = 0; r < 8; ++r) rmn[mi][r] = FLT_BIG;

#pragma unroll
    for (int mi = 0; mi < 2; ++mi) {
#pragma unroll
      for (int ni = 0; ni < 4; ++ni) {
        const int j = nsup + ni * 16 + l16;
        const float yyj = yysub[j];
        float cmin = FLT_BIG;
#pragma unroll
        for (int r = 0; r < 8; ++r) {
          const float v = yyj - 2.f * acc[mi][ni][r];
          cmin = fminf(cmin, v + xv[mi][r]);
          rmn[mi][r] = fminf(rmn[mi][r], v);
        }
        atomicMin(&colminU[j], fflip(cmin));
      }
#pragma unroll
      for (int r = 0; r < 8; ++r)
        atomicMin(&rowminU[msup + mi * 16 + 8 * half + r], fflip(rmn[mi][r] + xv[mi][r]));
    }
  }

  // ---- Pass 2: fused classifier heads O = Xblk @ M^T (64 x 400), M is n-major ----
  // 2-way M register blocking: each B fragment feeds 2 WMMAs
  for (int t = wave; t < 50; t += 8) {    // 2 M-supertiles (32 rows) x 25 N-tiles
    const int mtp = t / 25;               // 0..1 -> rows mtp*32 .. +31
    const int nt  = t % 25;
    const int n   = nt * 16 + l16;
    v8f acc0 = vzero, acc1 = vzero;
    const float* ap0 = Xs + (size_t)(mtp * 32 + l16) * LPAD + 2 * half;
    const float* ap1 = Xs + (size_t)(mtp * 32 + 16 + l16) * LPAD + 2 * half;
    const float* bp  = Mm + (size_t)n * LATENT + 2 * half;
    for (int kb = 0; kb < LATENT; kb += 4) {
      v2f a0 = *(const v2f*)(ap0 + kb);
      v2f a1 = *(const v2f*)(ap1 + kb);
      v2f b  = *(const v2f*)(bp + kb);
      acc0 = __builtin_amdgcn_wmma_f32_16x16x4_f32(false, a0, false, b, (short)0, acc0, false, false);
      acc1 = __builtin_amdgcn_wmma_f32_16x16x4_f32(false, a1, false, b, (short)0, acc1, false, false);
    }
    const float sv = svec[n];
    const float cv = cvec[n];
#pragma unroll
    for (int mi = 0; mi < 2; ++mi) {
#pragma unroll
      for (int r = 0; r < 8; ++r) {
        const int ml = mtp * 32 + mi * 16 + 8 * half + r;
        const size_t i = (size_t)(row0 + ml);
        const float a = (mi == 0) ? acc0[r] : acc1[r];
        const float val = xxs[ml] * sv + cv - 2.f * a;
        if (n < OUTN) out_sup[i * OUTN + n] = val;
        else          out_sub[i * OUTN + (n - OUTN)] = val;
      }
    }
  }

  __syncthreads();
  // block row-min sum -> global r2 partial
  if (tid < RB) xxs[tid] = funflip(rowminU[tid]);
  __syncthreads();
  for (int s = RB / 2; s > 0; s >>= 1) {
    if (tid < s) xxs[tid] += xxs[tid + s];
    __syncthreads();
  }
  if (tid == 0) atomicAdd(g_r2part, xxs[0]);
  // merge block col-mins into global
  for (int j = tid; j < NSUB; j += 256) atomicMin(&g_colmin[j], colminU[j]);
}

// ---- finale: r1 from global colmin, r2 from partial sum ----
__global__ void k_final(const unsigned* __restrict__ g_colmin,
                        const float* __restrict__ g_r2part, float* __restrict__ r_out) {
  __shared__ float red[1024];
  int t = threadIdx.x;
  red[t] = funflip(g_colmin[t]);
  __syncthreads();
  for (int s = 512; s > 0; s >>= 1) {
    if (t < s) red[t] += red[t + s];
    __syncthreads();
  }
  if (t == 0) {
    r_out[0] = red[0] / (float)NSUB;
    r_out[1] = *g_r2part / (float)BATCH;
  }
}

extern "C" void kernel_launch(void* const* d_in, const int* in_sizes, int n_in,
                              void* d_out, int out_size, void* d_ws, size_t ws_size,
                              hipStream_t stream) {
  (void)in_sizes; (void)n_in; (void)out_size; (void)ws_size;
  const float* x   = (const float*)d_in[0];
  const float* sup = (const float*)d_in[1];
  const float* sub = (const float*)d_in[2];
  const float* W1  = (const float*)d_in[3];
  const float* b1  = (const float*)d_in[4];
  const float* W2  = (const float*)d_in[5];
  const float* b2  = (const float*)d_in[6];

  float* out     = (float*)d_out;
  float* out_sub = out;
  float* out_sup = out + (size_t)BATCH * OUTN;
  float* r_out   = out + 2 * (size_t)BATCH * OUTN;   // r1,r2,r3,r4

  float* ws = (float*)d_ws;
  float*    yysub   = ws;                                // 1024
  float*    yysup   = ws + 1024;                         // 256
  float*    Mm      = ws + 1280;                         // 400*512 (n-major)
  float*    svec    = ws + 1280 + LATENT * NC;           // 400
  float*    cvec    = svec + NC;                         // 400
  unsigned* colminU = (unsigned*)(cvec + NC);            // 1024
  float*    r2part  = (float*)(colminU + NSUB);          // 1
  float*    Dss     = r2part + 4;                        // 256*1024

  k_row_sumsq<<<(NSUP + NSUB + 255) / 256, 256, 0, stream>>>(sup, sub, yysup, yysub);
  k_prep<<<(NSUB + 255) / 256, 256, 0, stream>>>(colminU, r2part);
  k_svec_cvec<<<(NC + 255) / 256, 256, 0, stream>>>(W1, b1, W2, b2, yysup, yysub, svec, cvec);
  k_build_M<<<(LATENT * NC + 255) / 256, 256, 0, stream>>>(sup, sub, W1, W2, Mm);
  k_dss<<<(NSUP * NSUB + 255) / 256, 256, 0, stream>>>(sup, sub, yysup, yysub, Dss);
  k_r3<<<1, 256, 0, stream>>>(Dss, r_out + 2);
  k_r4<<<1, 1024, 0, stream>>>(Dss, r_out + 3);

  const int smem = (RB * LPAD + RB + RB + NSUB) * (int)sizeof(float);
  hipFuncSetAttribute((const void*)k_main, hipFuncAttributeMaxDynamicSharedMemorySize, smem);
  k_main<<<BATCH / RB, 256, smem, stream>>>(x, sub, yysub, Mm, svec, cvec,
                                            out_sub, out_sup, colminU, r2part);
  k_final<<<1, 1024, 0, stream>>>(colminU, r2part, r_out);
}